// TFF_Point_SpatiallyAware_82910048682140
// MI455X (gfx1250) — compile-verified
//
#include <hip/hip_runtime.h>
#include <math.h>

typedef float v2f __attribute__((ext_vector_type(2)));
typedef float v8f __attribute__((ext_vector_type(8)));
typedef unsigned int u32x4 __attribute__((ext_vector_type(4)));
typedef int i32x8 __attribute__((ext_vector_type(8)));
typedef int i32x4 __attribute__((ext_vector_type(4)));

#define B_DIM 16
#define N_DIM 4096
#define M_DIM 1024
#define C_DIM 256

__device__ __forceinline__ float telu_act(float t) {
    // t * tanh(exp(t)); exp overflow -> tanh(inf)=1 -> t  (correct limit)
    return t * tanhf(__expf(t));
}

// ---------------------------------------------------------------------------
// Kernel 1: fused 3-NN (over M=1024 known points, staged in LDS) + inverse
// distance weighted interpolation of featB [B,C,M] -> interp [B,C,N].
// Thread-per-output-point; featB rows (4KB each, 16MB total) live in L2.
// ---------------------------------------------------------------------------
__global__ __launch_bounds__(128)
void three_nn_interp_kernel(const float* __restrict__ xyzA,
                            const float* __restrict__ xyzB,
                            const float* __restrict__ featB,
                            float* __restrict__ interp)
{
    __shared__ float sx[M_DIM], sy[M_DIM], sz[M_DIM];
    const int b = blockIdx.y;
    const int t = threadIdx.x;

    const float* xb = xyzB + (size_t)b * M_DIM * 3;
    for (int m = t; m < M_DIM; m += 128) {
        sx[m] = xb[m * 3 + 0];
        sy[m] = xb[m * 3 + 1];
        sz[m] = xb[m * 3 + 2];
    }
    __syncthreads();

    const int n = blockIdx.x * 128 + t;
    const float* xa = xyzA + ((size_t)b * N_DIM + n) * 3;
    const float ax = xa[0], ay = xa[1], az = xa[2];

    float d0 = 3.0e38f, d1 = 3.0e38f, d2 = 3.0e38f;
    int   i0 = 0, i1 = 0, i2 = 0;
    for (int m = 0; m < M_DIM; ++m) {
        float dx = ax - sx[m], dy = ay - sy[m], dz = az - sz[m];
        float d = dx * dx + dy * dy + dz * dz;
        if (d < d2) {
            if (d < d1) {
                if (d < d0) { d2 = d1; i2 = i1; d1 = d0; i1 = i0; d0 = d; i0 = m; }
                else        { d2 = d1; i2 = i1; d1 = d;  i1 = m; }
            } else          { d2 = d;  i2 = m; }
        }
    }
    float r0 = 1.f / (d0 + 1e-8f), r1 = 1.f / (d1 + 1e-8f), r2 = 1.f / (d2 + 1e-8f);
    float rs = 1.f / (r0 + r1 + r2);
    float w0 = r0 * rs, w1 = r1 * rs, w2 = r2 * rs;

    const float* fb = featB + (size_t)b * C_DIM * M_DIM;
    float* op = interp + (size_t)b * C_DIM * N_DIM + n;
    for (int c = 0; c < C_DIM; ++c) {
        const float* row = fb + (size_t)c * M_DIM;
        op[(size_t)c * N_DIM] = w0 * row[i0] + w1 * row[i1] + w2 * row[i2];
    }
}

// ---------------------------------------------------------------------------
// Fused WMMA GEMM:  Y[256 x 4096] = W[256 x K] * X[K x 4096]  per batch,
// using V_WMMA_F32_16X16X4_F32 (fp32-exact, memory-bound workload).
// Weight tile is fetched by the Tensor Data Mover (TENSOR_LOAD_TO_LDS) with
// TDM LDS-padding reproducing the bank-conflict-free pitch-36 layout.
//
// MODE 0: X = interp                        ; epilogue relu(BN)   -> outp (featB_on_A)
// MODE 1: X = [featA - fBA ; featA]        ; epilogue convA_w . telu(BN) -> atomic dotp
// MODE 2: X = [featA - fBA ; fBA]          ; epilogue convB_w . telu(BN) -> atomic dotp
// MODE 3: X = [Aw*featA ; Bw*fBA]          ; epilogue telu(BN)   -> outp (final x)
//
// Block tile 128(M) x 128(N), 8 waves of 32; wave tile 64 x 32 = 4x2 WMMA
// subtiles. K staged in LDS in chunks of 32.
// ---------------------------------------------------------------------------
template <int MODE>
__global__ __launch_bounds__(256)
void gemm_fused(const float* __restrict__ W,
                const float* __restrict__ gamma, const float* __restrict__ beta,
                const float* __restrict__ P,   // featA or interp
                const float* __restrict__ Q,   // featB_on_A
                const float* __restrict__ va,  // MODE1/2: conv vector [256]; MODE3: Aw [B,N]
                const float* __restrict__ vb,  // MODE3: Bw [B,N]
                float* __restrict__ outp,      // MODE0/3 output [B,256,N]
                float* __restrict__ dotp,      // MODE1/2 atomic accumulator [B,N]
                int K)
{
    constexpr int BM = 128, BN = 128, KC = 32;
    __shared__ float As[BM][KC + 4];   // 18.0 KB, row pitch 144B (16B aligned)
    __shared__ float Xs[KC][BN + 4];   // 16.5 KB, row pitch 528B (16B aligned)
    __shared__ float dotLds[BN];

    const int b   = blockIdx.z;
    const int n0  = blockIdx.x * BN;
    const int m0  = blockIdx.y * BM;
    const int t   = threadIdx.x;
    const int lane = t & 31;
    const int wv   = t >> 5;
    const int wm   = (wv & 1) * 64;    // wave M offset in block tile
    const int wn   = (wv >> 1) * 32;   // wave N offset in block tile
    const int hsel = lane >> 4;        // half-wave select (K-pair 0/1 vs 2/3)
    const int l16  = lane & 15;

    if (MODE == 1 || MODE == 2) { if (t < BN) dotLds[t] = 0.f; }

    v8f acc[4][2] = {};

    const size_t plane = (size_t)b * C_DIM * (size_t)N_DIM;

    for (int kc = 0; kc < K; kc += KC) {
        // ---- stage W chunk via TDM: 128 rows x 32 floats, stride K floats.
        // LDS padding: 4 DWORDs every 32 DWORDs -> matches As pitch of 36.
#if defined(__HIP_DEVICE_COMPILE__)
        if (wv == 0) {
            const uint64_t ga  = (uint64_t)(uintptr_t)(W + (size_t)m0 * K + kc);
            const uint32_t lds = (uint32_t)(uintptr_t)&As[0][0];
            u32x4 g0;
            g0[0] = 1u;                                   // count=1 (valid user D#)
            g0[1] = lds;                                  // lds_addr
            g0[2] = (uint32_t)ga;                         // global_addr[31:0]
            g0[3] = (uint32_t)((ga >> 32) & 0x01FFFFFFull) | (2u << 30); // addr hi | type=2
            i32x8 g1;
            g1[0] = (int)((2u << 16)      // data_size = 4 bytes
                        | (1u << 20)      // pad_enable
                        | (4u << 22)      // pad_interval: every 32 DWORDs
                        | (3u << 25));    // pad_amount: 4 DWORDs
            g1[1] = (int)(((uint32_t)K & 0xFFFFu) << 16);             // tensor_dim0 lo
            g1[2] = (int)(((uint32_t)K >> 16) | ((uint32_t)C_DIM << 16)); // dim0 hi | dim1 lo
            g1[3] = (int)((uint32_t)KC << 16);                        // dim1 hi | tile_dim0=32
            g1[4] = (int)BM;                                          // tile_dim1=128 | tile_dim2=0
            g1[5] = K;                                                // tensor_dim0_stride lo
            g1[6] = 0;                                                // stride0 hi | stride1 lo
            g1[7] = 0;                                                // stride1 hi
            i32x4 z4 = {};
            i32x8 z8 = {};
            __builtin_amdgcn_tensor_load_to_lds(g0, g1, z4, z4, z8, 0);
            __builtin_amdgcn_s_wait_tensorcnt(0);
        }
#else
        // host-pass placeholder (never codegen'd for device)
        if (wv == 0) {
            for (int i = t; i < BM * KC; i += 32)
                As[i / KC][i % KC] = W[(size_t)(m0 + i / KC) * K + kc + (i % KC)];
        }
#endif
        // ---- stage X chunk: 32 x 128 floats, formed on the fly ----
        #pragma unroll
        for (int i = 0; i < 4; ++i) {
            int idx = t + i * 256;
            int row = idx >> 5;               // 32 float4 per row
            int col = (idx & 31) << 2;
            int kg  = kc + row;
            int n   = n0 + col;
            float4 x4;
            if (MODE == 0) {
                x4 = *(const float4*)(P + plane + (size_t)kg * N_DIM + n);
            } else if (MODE == 1 || MODE == 2) {
                if (kg < C_DIM) {
                    float4 p4 = *(const float4*)(P + plane + (size_t)kg * N_DIM + n);
                    float4 q4 = *(const float4*)(Q + plane + (size_t)kg * N_DIM + n);
                    x4.x = p4.x - q4.x; x4.y = p4.y - q4.y;
                    x4.z = p4.z - q4.z; x4.w = p4.w - q4.w;
                } else {
                    const float* src = (MODE == 1) ? P : Q;
                    x4 = *(const float4*)(src + plane + (size_t)(kg - C_DIM) * N_DIM + n);
                }
            } else { // MODE 3
                if (kg < C_DIM) {
                    float4 p4 = *(const float4*)(P + plane + (size_t)kg * N_DIM + n);
                    float4 a4 = *(const float4*)(va + (size_t)b * N_DIM + n);
                    x4.x = a4.x * p4.x; x4.y = a4.y * p4.y;
                    x4.z = a4.z * p4.z; x4.w = a4.w * p4.w;
                } else {
                    float4 q4 = *(const float4*)(Q + plane + (size_t)(kg - C_DIM) * N_DIM + n);
                    float4 b4 = *(const float4*)(vb + (size_t)b * N_DIM + n);
                    x4.x = b4.x * q4.x; x4.y = b4.y * q4.y;
                    x4.z = b4.z * q4.z; x4.w = b4.w * q4.w;
                }
            }
            *(float4*)&Xs[row][col] = x4;
        }
        __syncthreads();

        // ---- WMMA over the K chunk, 4 at a time ----
        #pragma unroll
        for (int ks = 0; ks < KC; ks += 4) {
            v2f bfrag[2];
            #pragma unroll
            for (int nt = 0; nt < 2; ++nt) {
                int bn = wn + nt * 16 + l16;
                bfrag[nt].x = Xs[ks + hsel * 2 + 0][bn];
                bfrag[nt].y = Xs[ks + hsel * 2 + 1][bn];
            }
            #pragma unroll
            for (int mt = 0; mt < 4; ++mt) {
                int am = wm + mt * 16 + l16;
                v2f afrag;
                afrag.x = As[am][ks + hsel * 2 + 0];
                afrag.y = As[am][ks + hsel * 2 + 1];
                #pragma unroll
                for (int nt = 0; nt < 2; ++nt) {
                    acc[mt][nt] = __builtin_amdgcn_wmma_f32_16x16x4_f32(
                        false, afrag, false, bfrag[nt],
                        (short)0, acc[mt][nt], false, false);
                }
            }
        }
        __syncthreads();
    }

    const float inv = rsqrtf(1.f + 1e-5f);   // BN: running_var = 1

    if (MODE == 0 || MODE == 3) {
        float* op = outp + plane;
        #pragma unroll
        for (int mt = 0; mt < 4; ++mt) {
            #pragma unroll
            for (int r = 0; r < 8; ++r) {
                int m = m0 + wm + mt * 16 + hsel * 8 + r;
                float sc = gamma[m] * inv;
                float bb = beta[m];
                #pragma unroll
                for (int nt = 0; nt < 2; ++nt) {
                    int n = n0 + wn + nt * 16 + l16;
                    float tt = fmaf(acc[mt][nt][r], sc, bb);
                    float y = (MODE == 0) ? fmaxf(tt, 0.f) : telu_act(tt);
                    op[(size_t)m * N_DIM + n] = y;
                }
            }
        }
    } else {
        // Fused 1xC conv dot: reduce convW[m] * telu(bn(y[m,n])) over this
        // block's 128 m-rows, then atomically accumulate into dotp[b,n].
        float psum[2] = {0.f, 0.f};
        #pragma unroll
        for (int mt = 0; mt < 4; ++mt) {
            #pragma unroll
            for (int r = 0; r < 8; ++r) {
                int m = m0 + wm + mt * 16 + hsel * 8 + r;
                float sc = gamma[m] * inv;
                float bb = beta[m];
                float cw = va[m];
                #pragma unroll
                for (int nt = 0; nt < 2; ++nt) {
                    psum[nt] += cw * telu_act(fmaf(acc[mt][nt][r], sc, bb));
                }
            }
        }
        // lanes l and l+16 hold the same n (different m halves): combine.
        #pragma unroll
        for (int nt = 0; nt < 2; ++nt) {
            psum[nt] += __shfl_xor(psum[nt], 16, 32);
        }
        if (hsel == 0) {
            #pragma unroll
            for (int nt = 0; nt < 2; ++nt) {
                atomicAdd(&dotLds[wn + nt * 16 + l16], psum[nt]);
            }
        }
        __syncthreads();
        if (t < BN) {
            atomicAdd(dotp + (size_t)b * N_DIM + n0 + t, dotLds[t]);
        }
    }
}

// ---------------------------------------------------------------------------
// Small helpers
// ---------------------------------------------------------------------------
__global__ void zero_kernel(float* __restrict__ p, int n) {
    int i = blockIdx.x * 256 + threadIdx.x;
    if (i < n) p[i] = 0.f;
}

__global__ void sigmoid_kernel(float* __restrict__ aw, float* __restrict__ bw,
                               const float* __restrict__ ab,
                               const float* __restrict__ bb, int n) {
    int i = blockIdx.x * 256 + threadIdx.x;
    if (i < n) {
        aw[i] = 1.f / (1.f + __expf(-(aw[i] + ab[0])));
        bw[i] = 1.f / (1.f + __expf(-(bw[i] + bb[0])));
    }
}

// ---------------------------------------------------------------------------
extern "C" void kernel_launch(void* const* d_in, const int* in_sizes, int n_in,
                              void* d_out, int out_size, void* d_ws, size_t ws_size,
                              hipStream_t stream) {
    const float* xyzA   = (const float*)d_in[0];
    const float* featA  = (const float*)d_in[1];
    const float* xyzB   = (const float*)d_in[2];
    const float* featB  = (const float*)d_in[3];
    const float* fp_w   = (const float*)d_in[4];
    const float* fp_g   = (const float*)d_in[5];
    const float* fp_b   = (const float*)d_in[6];
    const float* catA_w = (const float*)d_in[7];
    const float* catA_g = (const float*)d_in[8];
    const float* catA_b = (const float*)d_in[9];
    const float* catB_w = (const float*)d_in[10];
    const float* catB_g = (const float*)d_in[11];
    const float* catB_b = (const float*)d_in[12];
    const float* cat_w  = (const float*)d_in[13];
    const float* cat_g  = (const float*)d_in[14];
    const float* cat_b  = (const float*)d_in[15];
    const float* convA_w = (const float*)d_in[16];
    const float* convA_b = (const float*)d_in[17];
    const float* convB_w = (const float*)d_in[18];
    const float* convB_b = (const float*)d_in[19];
    (void)in_sizes; (void)n_in; (void)out_size; (void)ws_size;

    float* out = (float*)d_out;
    float* ws  = (float*)d_ws;

    const size_t plane = (size_t)B_DIM * C_DIM * N_DIM;   // 16.78M floats
    float* interp = ws;                   // [B,C,N]
    float* fBA    = ws + plane;           // [B,C,N] featB_on_A
    float* awr    = ws + 2 * plane;       // [B,N]
    float* bwr    = awr + (size_t)B_DIM * N_DIM;

    const int bn = B_DIM * N_DIM;
    zero_kernel<<<(2 * bn + 255) / 256, 256, 0, stream>>>(awr, 2 * bn);

    three_nn_interp_kernel<<<dim3(N_DIM / 128, B_DIM), 128, 0, stream>>>(
        xyzA, xyzB, featB, interp);

    dim3 gg(N_DIM / 128, C_DIM / 128, B_DIM);   // (32, 2, 16)

    // G1: featB_on_A = relu(bn(fp_w @ interp))
    gemm_fused<0><<<gg, 256, 0, stream>>>(fp_w, fp_g, fp_b,
                                          interp, nullptr, nullptr, nullptr,
                                          fBA, nullptr, C_DIM);
    // G2: Aw_raw = convA_w . telu(bn(catA_w @ [featA - fBA ; featA]))
    gemm_fused<1><<<gg, 256, 0, stream>>>(catA_w, catA_g, catA_b,
                                          featA, fBA, convA_w, nullptr,
                                          nullptr, awr, 2 * C_DIM);
    // G3: Bw_raw = convB_w . telu(bn(catB_w @ [featA - fBA ; fBA]))
    gemm_fused<2><<<gg, 256, 0, stream>>>(catB_w, catB_g, catB_b,
                                          featA, fBA, convB_w, nullptr,
                                          nullptr, bwr, 2 * C_DIM);

    sigmoid_kernel<<<(bn + 255) / 256, 256, 0, stream>>>(awr, bwr, convA_b, convB_b, bn);

    // G5: x = telu(bn(cat_w @ [Aw*featA ; Bw*fBA]))
    gemm_fused<3><<<gg, 256, 0, stream>>>(cat_w, cat_g, cat_b,
                                          featA, fBA, awr, bwr,
                                          out, nullptr, 2 * C_DIM);
}